// Yolov1Loss_51711406244499
// MI455X (gfx1250) — compile-verified
//
#include <hip/hip_runtime.h>
#include <stdint.h>

typedef __attribute__((ext_vector_type(2))) float v2f;
typedef __attribute__((ext_vector_type(8))) float v8f;

#define TPB 256
#define NBLK 784
#define TILES_PER_BLK 4
#define NWAVES (TPB / 32)

constexpr int   kNC    = 20;
constexpr int   kGS    = 7;
constexpr int   kCh    = 30;                         // channels per cell
constexpr int   kBatch = 16384;
constexpr int   kCells = kBatch * kGS * kGS;         // 802816 = 784*4*256
constexpr float kLC    = 5.0f;                       // LAMBDA_COORD
constexpr float kLN    = 0.5f;                       // LAMBDA_NOOBJ

constexpr unsigned kTileCells   = TPB;                       // 256 cells / tile
constexpr unsigned kTileBytes   = kTileCells * kCh * 4u;     // 30720 B per array
constexpr unsigned kTileXfers   = kTileBytes / 16u;          // 1920 b128 moves
constexpr unsigned kBufBytes    = 2u * kTileBytes;           // preds+targets
constexpr unsigned kBufFloats   = kBufBytes / 4u;            // 15360
constexpr unsigned kTileFloats  = kTileBytes / 4u;           // 7680
constexpr unsigned kSmemBytes   = 2u * kBufBytes;            // 122880 (double buf)

// ---------------------------------------------------------------------------
// Wave-level sum across 32 lanes via V_WMMA_F32_16X16X4_F32.
//   A(16x4): lane m -> A[m][0]=v, A[m][1]=0; lane m+16 -> A[m][2]=v, A[m][3]=0
//   B = ones (layout-invariant)  =>  D[m][n] = v[m] + v[m+16]
// Sum the 8 D VGPRs (rows 0..7 in lanes 0..15, rows 8..15 in lanes 16..31),
// fold the halves with shfl_xor(16).
// ---------------------------------------------------------------------------
__device__ __forceinline__ float wave_sum32(float v) {
  v2f a; a[0] = v;    a[1] = 0.0f;
  v2f b; b[0] = 1.0f; b[1] = 1.0f;
  v8f c = {};
  v8f d = __builtin_amdgcn_wmma_f32_16x16x4_f32(false, a, false, b,
                                                (short)0, c, false, false);
  float s = d[0] + d[1] + d[2] + d[3] + d[4] + d[5] + d[6] + d[7];
  s += __shfl_xor(s, 16, 32);
  return s;
}

__device__ __forceinline__ void block_sum3(float g0, float g1, float g2,
                                           float* __restrict__ out3) {
  __shared__ float lds[NWAVES][3];
  float w0 = wave_sum32(g0);
  float w1 = wave_sum32(g1);
  float w2 = wave_sum32(g2);
  const int lane = threadIdx.x & 31;
  const int wave = threadIdx.x >> 5;
  if (lane == 0) { lds[wave][0] = w0; lds[wave][1] = w1; lds[wave][2] = w2; }
  __syncthreads();
  if (threadIdx.x < 3) {
    float s = 0.0f;
#pragma unroll
    for (int w = 0; w < NWAVES; ++w) s += lds[w][threadIdx.x];
    out3[threadIdx.x] = s;
  }
}

// ---------------------------------------------------------------------------
// Async-stage one tile (preds+targets) into an LDS buffer with coalesced
// b128 global->LDS DMA. 1920 moves/array; threads do 7 each, tids<128 an 8th
// (wave-uniform predicate: boundary is at wave 4). Tracked by ASYNCcnt.
// ---------------------------------------------------------------------------
__device__ __forceinline__ void stage_tile(const float* __restrict__ preds,
                                           const float* __restrict__ targets,
                                           unsigned ldsBufByte, unsigned tile,
                                           unsigned tid) {
  const unsigned gbase = tile * kTileBytes;        // byte offset within array
  const unsigned ldsP  = ldsBufByte;
  const unsigned ldsT  = ldsBufByte + kTileBytes;
#pragma unroll
  for (unsigned j = 0; j < 7; ++j) {
    const unsigned o = (tid + 256u * j) * 16u;
    asm volatile("global_load_async_to_lds_b128 %0, %1, %2"
                 :: "v"(ldsP + o), "v"(gbase + o), "s"(preds) : "memory");
    asm volatile("global_load_async_to_lds_b128 %0, %1, %2"
                 :: "v"(ldsT + o), "v"(gbase + o), "s"(targets) : "memory");
  }
  if (tid < 128u) {                                // waves 0..3 only (uniform)
    const unsigned o = (tid + 1792u) * 16u;
    asm volatile("global_load_async_to_lds_b128 %0, %1, %2"
                 :: "v"(ldsP + o), "v"(gbase + o), "s"(preds) : "memory");
    asm volatile("global_load_async_to_lds_b128 %0, %1, %2"
                 :: "v"(ldsT + o), "v"(gbase + o), "s"(targets) : "memory");
  }
}

// ---------------------------------------------------------------------------
// Kernel 1: double-buffered async pipeline. Per tile: wait ASYNCcnt==0
// (only the current tile's loads are outstanding at that point), barrier,
// kick off next tile's DMA, compute this tile from LDS.
// ---------------------------------------------------------------------------
__global__ __launch_bounds__(TPB)
void yolo_partial(const float* __restrict__ preds,
                  const float* __restrict__ targets,
                  float* __restrict__ partial) {
  extern __shared__ float smem[];
  const unsigned tid     = threadIdx.x;
  const unsigned ldsBase = (unsigned)(uintptr_t)smem;
  const unsigned tile0   = blockIdx.x * TILES_PER_BLK;
  const float    inv     = 1.0f / (float)kGS;

  float g0 = 0.0f, g1 = 0.0f, g2 = 0.0f;

  stage_tile(preds, targets, ldsBase, tile0, tid);

  for (int k = 0; k < TILES_PER_BLK; ++k) {
    asm volatile("s_wait_asynccnt 0x0" ::: "memory");  // my tile-k loads done
    __syncthreads();                                   // everyone's are done
    if (k + 1 < TILES_PER_BLK)
      stage_tile(preds, targets, ldsBase + (unsigned)((k + 1) & 1) * kBufBytes,
                 tile0 + (unsigned)(k + 1), tid);      // overlaps compute

    const float* P = smem + (unsigned)(k & 1) * kBufFloats + tid * kCh;
    const float* T = P + kTileFloats;

    float p[kCh], t[kCh];
    const v2f* pp = (const v2f*)P;
    const v2f* tp = (const v2f*)T;
#pragma unroll
    for (int i = 0; i < kCh / 2; ++i) { v2f v = pp[i]; p[2*i] = v[0]; p[2*i+1] = v[1]; }
#pragma unroll
    for (int i = 0; i < kCh / 2; ++i) { v2f v = tp[i]; t[2*i] = v[0]; t[2*i+1] = v[1]; }

    const bool  obj = t[4] > 0.0f;
    const float m   = obj ? 1.0f : 0.0f;

    // target box (box 0) in xyxy
    const float bx0 = t[0] * inv - 0.5f * t[2];
    const float by0 = t[1] * inv - 0.5f * t[3];
    const float bx1 = t[0] * inv + 0.5f * t[2];
    const float by1 = t[1] * inv + 0.5f * t[3];
    const float areaB = (bx1 - bx0) * (by1 - by0);

    float iou0, iou1;
#pragma unroll
    for (int b = 0; b < 2; ++b) {
      const float px = p[5*b+0], py = p[5*b+1], pw = p[5*b+2], ph = p[5*b+3];
      const float ax0 = px * inv - 0.5f * pw;
      const float ay0 = py * inv - 0.5f * ph;
      const float ax1 = px * inv + 0.5f * pw;
      const float ay1 = py * inv + 0.5f * ph;
      const float areaA = (ax1 - ax0) * (ay1 - ay0);
      float iw = fminf(ax1, bx1) - fmaxf(ax0, bx0);
      float ih = fminf(ay1, by1) - fmaxf(ay0, by0);
      iw = fmaxf(iw, 0.0f); ih = fmaxf(ih, 0.0f);
      const float inter = iw * ih;
      const float r = inter / (areaA + areaB - inter);
      if (b == 0) iou0 = r; else iou1 = r;
    }

    const bool  r1     = iou1 > iou0;                 // jnp.argmax tie -> 0
    const float maxIou = fmaxf(iou0, iou1);
    const float rp0 = r1 ? p[5] : p[0];
    const float rp1 = r1 ? p[6] : p[1];
    const float rp2 = r1 ? p[7] : p[2];
    const float rp3 = r1 ? p[8] : p[3];
    const float rp4 = r1 ? p[9] : p[4];

    const float dx = rp0 - t[0], dy = rp1 - t[1];
    const float sp0 = sqrtf(obj ? rp2 : 1.0f);
    const float sp1 = sqrtf(obj ? rp3 : 1.0f);
    const float st0 = sqrtf(obj ? t[2] : 1.0f);
    const float st1 = sqrtf(obj ? t[3] : 1.0f);
    const float dw = sp0 - st0, dh = sp1 - st1;
    g0 += m * (dx * dx + dy * dy) + (obj ? (dw * dw + dh * dh) : 0.0f);

    const float dob = rp4 - maxIou;
    const float n0 = p[4] - t[4], n1 = p[9] - t[9];
    g1 += m * dob * dob + (obj ? 0.0f : kLN * (n0 * n0 + n1 * n1));

    float cs = 0.0f;
#pragma unroll
    for (int c = 0; c < kNC; ++c) {
      const float d = p[10 + c] - t[10 + c];
      cs += d * d;
    }
    g2 += m * cs;
  }

  block_sum3(g0, g1, g2, partial + (size_t)blockIdx.x * 3);
}

// ---------------------------------------------------------------------------
// Kernel 2: one block deterministically reduces the NBLK partials and scales.
// ---------------------------------------------------------------------------
__global__ __launch_bounds__(TPB)
void yolo_final(const float* __restrict__ partial, float* __restrict__ out) {
  float g0 = 0.0f, g1 = 0.0f, g2 = 0.0f;
  for (int i = threadIdx.x; i < NBLK; i += TPB) {
    g0 += partial[3 * i + 0];
    g1 += partial[3 * i + 1];
    g2 += partial[3 * i + 2];
  }
  __shared__ float res[3];
  block_sum3(g0, g1, g2, res);
  __syncthreads();
  if (threadIdx.x == 0) {
    const float invBS = 1.0f / (float)kBatch;
    out[0] = kLC * res[0] * invBS;   // loss_boxes
    out[1] = res[1] * invBS;         // loss_object (noobj pre-weighted)
    out[2] = res[2] * invBS;         // loss_class
  }
}

extern "C" void kernel_launch(void* const* d_in, const int* in_sizes, int n_in,
                              void* d_out, int out_size, void* d_ws, size_t ws_size,
                              hipStream_t stream) {
  const float* preds   = (const float*)d_in[0];
  const float* targets = (const float*)d_in[1];
  float* out     = (float*)d_out;
  float* partial = (float*)d_ws;    // NBLK*3 floats, fully rewritten every call

  yolo_partial<<<NBLK, TPB, kSmemBytes, stream>>>(preds, targets, partial);
  yolo_final<<<1, TPB, 0, stream>>>(partial, out);
}